// MaskFiller_22428319220382
// MI455X (gfx1250) — compile-verified
//
#include <hip/hip_runtime.h>

// Problem constants from the reference (fixed by setup_inputs()):
//   B=8, K=4096, M=4096, D=512, L=K+M=8192, float32 everywhere.
constexpr int Bc = 8;
constexpr int Kc = 4096;
constexpr int Mc = 4096;
constexpr int Dc = 512;
constexpr int Lc = Kc + Mc;

// 128-bit vector type that clang's nontemporal builtins accept and that
// lowers to global_load_b128 / global_store_b128 on gfx1250.
typedef __attribute__((ext_vector_type(4))) float v4f;

// One block per kept input row: copy inputs[b,k,:] -> out[b, keep_ids[b,k], :].
// blockDim.x == D/4 == 128 lanes (4 wave32s); each lane moves one float4.
// Streamed data (read-once input, write-once output) uses non-temporal hints.
__global__ __launch_bounds__(Dc / 4) void maskfiller_scatter_keep(
    const float* __restrict__ in,        // [B*K*D]
    const int* __restrict__ keep_ids,    // [B*K]
    float* __restrict__ out)             // [B*L*D]
{
    const int r = blockIdx.x;            // r = b*K + k, uniform per block
    const int b = r / Kc;
    // Hint the (uniform) index load path; lowers to global_prefetch_b8.
    __builtin_prefetch(&keep_ids[r], 0, 0);
    const int dst = keep_ids[r];

    const v4f* __restrict__ src =
        reinterpret_cast<const v4f*>(in) + (size_t)r * (Dc / 4);
    v4f* __restrict__ dp =
        reinterpret_cast<v4f*>(out) + ((size_t)b * Lc + dst) * (Dc / 4);

    // Read-once / write-once 2KB row: keep it out of the caches (TH_NT).
    v4f v = __builtin_nontemporal_load(&src[threadIdx.x]);
    __builtin_nontemporal_store(v, &dp[threadIdx.x]);
}

// One block per masked row: broadcast mask_embedding -> out[b, mask_ids[b,m], :].
// The 2KB embedding is re-read by 32768 blocks, so load it with regular
// (temporal) loads to keep it resident in WGP$/L2; the output store is
// write-once and goes non-temporal.
__global__ __launch_bounds__(Dc / 4) void maskfiller_scatter_mask(
    const float* __restrict__ emb,       // [D]
    const int* __restrict__ mask_ids,    // [B*M]
    float* __restrict__ out)             // [B*L*D]
{
    const int r = blockIdx.x;            // r = b*M + m, uniform per block
    const int b = r / Mc;
    const int dst = mask_ids[r];

    const v4f* __restrict__ ep = reinterpret_cast<const v4f*>(emb);
    v4f* __restrict__ dp =
        reinterpret_cast<v4f*>(out) + ((size_t)b * Lc + dst) * (Dc / 4);

    v4f v = ep[threadIdx.x];             // temporal: stays hot in cache
    __builtin_nontemporal_store(v, &dp[threadIdx.x]);
}

extern "C" void kernel_launch(void* const* d_in, const int* in_sizes, int n_in,
                              void* d_out, int out_size, void* d_ws, size_t ws_size,
                              hipStream_t stream) {
    // setup_inputs() dict order:
    //   0: inputs            float32 [B*K*D]
    //   1: mask_position_ids int32   [B*M]
    //   2: keep_position_ids int32   [B*K]
    //   3: mask_embedding    float32 [D]
    //   4: axis              (scalar, ignored; reference asserts axis == -2)
    const float* in       = reinterpret_cast<const float*>(d_in[0]);
    const int*   mask_ids = reinterpret_cast<const int*>(d_in[1]);
    const int*   keep_ids = reinterpret_cast<const int*>(d_in[2]);
    const float* emb      = reinterpret_cast<const float*>(d_in[3]);
    float*       out      = reinterpret_cast<float*>(d_out);

    // keep/mask indices partition [0, L) per batch, so every output row is
    // written exactly once: no d_out initialization required, and the two
    // kernels touch disjoint rows (safe to launch back-to-back on stream).
    dim3 block(Dc / 4);                      // 128 threads = 4 wave32s
    maskfiller_scatter_keep<<<dim3(Bc * Kc), block, 0, stream>>>(in, keep_ids, out);
    maskfiller_scatter_mask<<<dim3(Bc * Mc), block, 0, stream>>>(emb, mask_ids, out);
}